// DriftingModel_55319178772707
// MI455X (gfx1250) — compile-verified
//
#include <hip/hip_runtime.h>
#include <hip/hip_bf16.h>
#include <math.h>

// Problem constants (match the reference)
#define NPTS   16384
#define NOISE  32
#define HID    256
#define DDIM   2
#define TEMP_F 0.05f

// LDS activation row stride: 256 + 4 pad words -> conflict-free A-fragment
// reads and conflict-free C/D stores.
#define LDS_STRIDE 260

typedef __attribute__((ext_vector_type(2))) float v2f;
typedef __attribute__((ext_vector_type(8))) float v8f;

__device__ __forceinline__ float selu_f(float x) {
    const float lam   = 1.0507009873554805f;
    const float alpha = 1.6732632423543772f;
    // exp(x) = exp2(x * log2(e)) -> single v_exp_f32
    float e = exp2f(x * 1.4426950408889634f);
    return (x > 0.0f) ? (lam * x) : (lam * alpha * (e - 1.0f));
}

// One hidden layer: dst = selu(src @ W + b), src/dst are this wave's 16x256
// activation slabs in LDS. K is marched 4 at a time with fp32 WMMA.
__device__ __forceinline__ void layer_wmma(const float* __restrict__ W,
                                           const float* __restrict__ bvec,
                                           const float* src, float* dst,
                                           int m, int half, int kb) {
#pragma unroll 1
    for (int t = 0; t < HID / 16; ++t) {
        const int n = t * 16 + m;
        const float bias = bvec[n];
        v8f acc;
#pragma unroll
        for (int r = 0; r < 8; ++r) acc[r] = bias;
#pragma unroll 8
        for (int k0 = 0; k0 < HID; k0 += 4) {
            v2f a, b;
            // A fragment: lane holds row m, K = k0+kb, k0+kb+1 (contiguous)
            a.x = src[m * LDS_STRIDE + k0 + kb];
            a.y = src[m * LDS_STRIDE + k0 + kb + 1];
            // B fragment: lane holds col n, K rows k0+kb and k0+kb+1
            b.x = W[(k0 + kb) * HID + n];
            b.y = W[(k0 + kb + 1) * HID + n];
            acc = __builtin_amdgcn_wmma_f32_16x16x4_f32(
                false, a, false, b, (short)0, acc, false, false);
        }
#pragma unroll
        for (int r = 0; r < 8; ++r) {
            const int row = r + half * 8;
            dst[row * LDS_STRIDE + n] = selu_f(acc[r]);
        }
    }
}

// One wave (32 threads) per block; each wave generates 16 rows of `gen`.
__global__ __launch_bounds__(32) void mlp_gen_kernel(
    const float* __restrict__ z,
    const float* __restrict__ W1, const float* __restrict__ b1,
    const float* __restrict__ W2, const float* __restrict__ b2,
    const float* __restrict__ W3, const float* __restrict__ b3,
    const float* __restrict__ W4, const float* __restrict__ b4,
    const float* __restrict__ W5, const float* __restrict__ b5,
    float* __restrict__ gen) {
    __shared__ float hA[16 * LDS_STRIDE];
    __shared__ float hB[16 * LDS_STRIDE];

    const int lane = threadIdx.x;   // wave32
    const int m    = lane & 15;     // row within 16-row tile / col within n-tile
    const int half = lane >> 4;     // which K half this lane covers
    const int kb   = half * 2;
    const int rowbase = blockIdx.x * 16;

    // ---- Layer 1: hA = selu(z @ W1 + b1), K = 32, A straight from global ----
#pragma unroll 1
    for (int t = 0; t < HID / 16; ++t) {
        const int n = t * 16 + m;
        const float bias = b1[n];
        v8f acc;
#pragma unroll
        for (int r = 0; r < 8; ++r) acc[r] = bias;
#pragma unroll
        for (int k0 = 0; k0 < NOISE; k0 += 4) {
            v2f a, b;
            const float* zp = z + (rowbase + m) * NOISE + k0 + kb;
            a.x = zp[0];
            a.y = zp[1];
            b.x = W1[(k0 + kb) * HID + n];
            b.y = W1[(k0 + kb + 1) * HID + n];
            acc = __builtin_amdgcn_wmma_f32_16x16x4_f32(
                false, a, false, b, (short)0, acc, false, false);
        }
#pragma unroll
        for (int r = 0; r < 8; ++r) {
            const int row = r + half * 8;
            hA[row * LDS_STRIDE + n] = selu_f(acc[r]);
        }
    }
    __syncthreads();

    // ---- Layers 2-4, ping-pong LDS slabs ----
    layer_wmma(W2, b2, hA, hB, m, half, kb);
    __syncthreads();
    layer_wmma(W3, b3, hB, hA, m, half, kb);
    __syncthreads();
    layer_wmma(W4, b4, hA, hB, m, half, kb);
    __syncthreads();

    // ---- Layer 5: gen = h4 @ W5 + b5 (HID -> 2). 32 lanes = 16 rows x 2 cols.
    {
        const float* h4 = hB;
        const int col = half;  // DDIM == 2
        float accv = b5[col];
#pragma unroll 8
        for (int k = 0; k < HID; k += 4) {
            float h0 = h4[m * LDS_STRIDE + k + 0];
            float h1 = h4[m * LDS_STRIDE + k + 1];
            float h2 = h4[m * LDS_STRIDE + k + 2];
            float h3 = h4[m * LDS_STRIDE + k + 3];
            accv = fmaf(h0, W5[(k + 0) * DDIM + col], accv);
            accv = fmaf(h1, W5[(k + 1) * DDIM + col], accv);
            accv = fmaf(h2, W5[(k + 2) * DDIM + col], accv);
            accv = fmaf(h3, W5[(k + 3) * DDIM + col], accv);
        }
        gen[(rowbase + m) * DDIM + col] = accv;
    }
}

// ---------------- Energy: chunked distance + streaming logsumexp ------------
// Column space split into NCHUNK chunks so the grid (64 x 8 blocks, 4096
// waves) fills every SIMD on the MI455X; a combine kernel merges the
// chunk-local shifted logsumexp partials exactly.
#define TILE   1024            // points staged per LDS tile (8 KB of float2)
#define NCHUNK 8
#define CHUNK  (NPTS / NCHUNK) // 2048 columns per block

#define K_EXP_C ((1.0f / TEMP_F) * 1.4426950408889634f)  // 1/T * log2(e)
#define T_LN2_C (TEMP_F * 0.6931471805599453f)           // T * ln(2)
#define BIG_C   3.4e38f

__global__ __launch_bounds__(256) void energy_partial_kernel(
    const float* __restrict__ pos, const float* __restrict__ gen,
    float* __restrict__ dmin_part, float* __restrict__ s_part) {
    __shared__ float2 tile[TILE];
    const int i     = blockIdx.x * blockDim.x + threadIdx.x;  // output row
    const int chunk = blockIdx.y;
    const int cbase = chunk * CHUNK;

    const float2 g = ((const float2*)gen)[i];
    const float2* srcs[2] = {(const float2*)pos, (const float2*)gen};

#pragma unroll
    for (int a = 0; a < 2; ++a) {       // a==0: pos sweep, a==1: gen sweep
        const float2* src = srcs[a];

        // ---- pass 1: chunk-local min squared distance ----
        float d2min = BIG_C;
        for (int base = cbase; base < cbase + CHUNK; base += TILE) {
            __syncthreads();
            for (int j = threadIdx.x; j < TILE; j += blockDim.x)
                tile[j] = src[base + j];
            __syncthreads();
            // warm the next tile (global_prefetch_b8) while we compute
            __builtin_prefetch(&src[base + TILE + threadIdx.x * 4]);
#pragma unroll 8
            for (int j = 0; j < TILE; ++j) {
                float dx = g.x - tile[j].x, dy = g.y - tile[j].y;
                float d2 = fmaf(dx, dx, dy * dy);
                bool diag = (a == 1) && (base + j == i);
                d2 = diag ? BIG_C : d2;
                d2min = fminf(d2min, d2);
            }
        }
        const float dmin = sqrtf(d2min);

        // ---- pass 2: chunk-local shifted exponential sum ----
        float s = 0.0f;
        for (int base = cbase; base < cbase + CHUNK; base += TILE) {
            __syncthreads();
            for (int j = threadIdx.x; j < TILE; j += blockDim.x)
                tile[j] = src[base + j];
            __syncthreads();
            __builtin_prefetch(&src[base + TILE + threadIdx.x * 4]);
#pragma unroll 4
            for (int j = 0; j < TILE; ++j) {
                float dx = g.x - tile[j].x, dy = g.y - tile[j].y;
                float d  = sqrtf(fmaf(dx, dx, dy * dy));
                float e  = exp2f((dmin - d) * K_EXP_C);
                bool diag = (a == 1) && (base + j == i);
                s += diag ? 0.0f : e;
            }
        }

        dmin_part[(a * NCHUNK + chunk) * NPTS + i] = dmin;
        s_part[(a * NCHUNK + chunk) * NPTS + i]    = s;
    }
}

__global__ __launch_bounds__(256) void energy_combine_kernel(
    const float* __restrict__ dmin_part, const float* __restrict__ s_part,
    float* __restrict__ out) {
    const int i = blockIdx.x * blockDim.x + threadIdx.x;

    float dglob[2], ssum[2];
#pragma unroll
    for (int a = 0; a < 2; ++a) {
        float dg = BIG_C;
#pragma unroll
        for (int c = 0; c < NCHUNK; ++c)
            dg = fminf(dg, dmin_part[(a * NCHUNK + c) * NPTS + i]);
        float s = 0.0f;
#pragma unroll
        for (int c = 0; c < NCHUNK; ++c) {
            float dc = dmin_part[(a * NCHUNK + c) * NPTS + i];
            float sc = s_part[(a * NCHUNK + c) * NPTS + i];
            s += sc * exp2f((dg - dc) * K_EXP_C);  // exact chunk merge
        }
        dglob[a] = dg;
        ssum[a]  = s;
    }
    // energy_pos = dmin_p - T*log(s_p) ; energy_neg = -dmin_g + T*log(s_g)
    out[i] = dglob[0] - T_LN2_C * log2f(ssum[0])
           - dglob[1] + T_LN2_C * log2f(ssum[1]);
}

extern "C" void kernel_launch(void* const* d_in, const int* in_sizes, int n_in,
                              void* d_out, int out_size, void* d_ws, size_t ws_size,
                              hipStream_t stream) {
    const float* pos = (const float*)d_in[0];
    const float* z   = (const float*)d_in[1];
    const float* W1  = (const float*)d_in[2];
    const float* b1  = (const float*)d_in[3];
    const float* W2  = (const float*)d_in[4];
    const float* b2  = (const float*)d_in[5];
    const float* W3  = (const float*)d_in[6];
    const float* b3  = (const float*)d_in[7];
    const float* W4  = (const float*)d_in[8];
    const float* b4  = (const float*)d_in[9];
    const float* W5  = (const float*)d_in[10];
    const float* b5  = (const float*)d_in[11];

    // workspace layout (floats):
    //   [0, NPTS*2)                       gen
    //   [NPTS*2, NPTS*2 + NPTS*2*NCHUNK)  dmin partials
    //   [...,    ... + NPTS*2*NCHUNK)     s partials        (~2.2 MB total)
    float* ws        = (float*)d_ws;
    float* gen       = ws;
    float* dmin_part = ws + (size_t)NPTS * 2;
    float* s_part    = dmin_part + (size_t)NPTS * 2 * NCHUNK;
    float* out       = (float*)d_out;

    mlp_gen_kernel<<<NPTS / 16, 32, 0, stream>>>(z, W1, b1, W2, b2, W3, b3,
                                                 W4, b4, W5, b5, gen);
    energy_partial_kernel<<<dim3(NPTS / 256, NCHUNK), 256, 0, stream>>>(
        pos, gen, dmin_part, s_part);
    energy_combine_kernel<<<NPTS / 256, 256, 0, stream>>>(dmin_part, s_part,
                                                          out);
}